// Detection_44848048505355
// MI455X (gfx1250) — compile-verified
//
#include <hip/hip_runtime.h>
#include <math.h>

#define BATCH 8
#define NANCH 4096
#define NCLS  5
#define OUTSTRIDE 24576   // 2*N + 4*N per batch row
#define KEEPOFF   8192
#define TPB   512
#define NWORDS (NANCH/32) // 128 bitmask words

typedef __attribute__((ext_vector_type(2))) float v2f;
typedef __attribute__((ext_vector_type(8))) float v8f;

// ---------------- kernel A: decode boxes into out, zero kept-score region ----
__global__ __launch_bounds__(256)
void decode_zero_kernel(const float* __restrict__ loc,
                        const float* __restrict__ defs,
                        float* __restrict__ out) {
  int g = blockIdx.x * blockDim.x + threadIdx.x;
  if (g >= BATCH * NANCH) return;
  int b = g >> 12, n = g & (NANCH - 1);
  float d0 = defs[2*n], d1 = defs[2*n+1];
  float x = loc[2*g], y = loc[2*g+1];
  float ctr = fmaf(x, d1, d0);
  float w   = d1 * expf(y);
  float* ob = out + (size_t)b * OUTSTRIDE;
  ob[2*n]   = ctr - 0.5f*w;
  ob[2*n+1] = ctr + 0.5f*w;
  ob[KEEPOFF + 0*NANCH + n] = 0.f;
  ob[KEEPOFF + 1*NANCH + n] = 0.f;
  ob[KEEPOFF + 2*NANCH + n] = 0.f;
  ob[KEEPOFF + 3*NANCH + n] = 0.f;
}

// softmax score of fg class c (0..3 -> logits index c+1)
__device__ __forceinline__ float fg_score(const float* z, int c) {
  float z0=z[0], z1=z[1], z2=z[2], z3=z[3], z4=z[4];
  float m = fmaxf(fmaxf(fmaxf(z0,z1), fmaxf(z2,z3)), z4);
  float e0=expf(z0-m), e1=expf(z1-m), e2=expf(z2-m), e3=expf(z3-m), e4=expf(z4-m);
  float denom = e0+e1+e2+e3+e4;
  float ec = (c==0)?e1:(c==1)?e2:(c==2)?e3:e4;
  return ec/denom;
}

// ---------------- kernel B: one workgroup per (batch, class) NMS -------------
// LDS layout (aliased; total 49664 B + 2 scalars):
//  [0,16K)      phase1: sort keys (f32)   / phase2: sorted starts sS (f32)
//  [16K,32K)    phase2: sorted ends  sE (f32)
//  [32K,40K)    sIdx (u16), persists
//  [40K,40.5K)  keepMask (128 u32)
//  [40.5K,48.5K)rowbits[16][128] (u32)
__global__ __launch_bounds__(TPB)
void nms_kernel(const float* __restrict__ loc, const float* __restrict__ cls,
                const float* __restrict__ defs, float* __restrict__ out) {
  __shared__ __align__(16) unsigned char smem[49664];
  __shared__ int sM;
  __shared__ unsigned sAct;

  float*          sKey     = (float*)smem;                     // phase 1
  float*          sS       = (float*)smem;                     // phase 2 (overlays keys)
  float*          sE       = (float*)(smem + 16384);
  unsigned short* sIdx     = (unsigned short*)(smem + 32768);
  unsigned*       keepMask = (unsigned*)(smem + 40960);
  unsigned (*rowbits)[NWORDS] = (unsigned(*)[NWORDS])(smem + 41472);

  const int tid = threadIdx.x;
  const int b   = blockIdx.x >> 2;
  const int c   = blockIdx.x & 3;

  if (tid == 0) sM = 0;
  for (int w = tid; w < NWORDS; w += TPB) keepMask[w] = 0xFFFFFFFFu;
  __syncthreads();

  // ---- phase 1: masked scores ----
  for (int p = tid; p < NANCH; p += TPB) {
    const float* z = cls + ((size_t)(b*NANCH + p)) * NCLS;
    float sc = fg_score(z, c);
    bool valid = sc > 0.5f;
    sKey[p] = valid ? sc : -1e30f;
    sIdx[p] = (unsigned short)p;
    if (valid) atomicAdd(&sM, 1);
  }
  __syncthreads();

  // ---- bitonic sort, descending by key ----
  for (int k = 2; k <= NANCH; k <<= 1) {
    for (int jj = k >> 1; jj > 0; jj >>= 1) {
      for (int i = tid; i < NANCH; i += TPB) {
        int ixj = i ^ jj;
        if (ixj > i) {
          bool desc = ((i & k) == 0);
          float a = sKey[i], bb = sKey[ixj];
          if (desc ? (a < bb) : (a > bb)) {
            sKey[i] = bb; sKey[ixj] = a;
            unsigned short t = sIdx[i]; sIdx[i] = sIdx[ixj]; sIdx[ixj] = t;
          }
        }
      }
      __syncthreads();
    }
  }
  const int M = sM;
  __syncthreads();  // keys dead after this point; region becomes sS

  // ---- gather sorted boxes (recompute decode; keys region reused) ----
  for (int p = tid; p < NANCH; p += TPB) {
    int o = (int)sIdx[p];
    const float* lp = loc + ((size_t)(b*NANCH) + o)*2;
    float d0 = defs[2*o], d1 = defs[2*o+1];
    float ctr = fmaf(lp[0], d1, d0);
    float w   = d1 * expf(lp[1]);
    sS[p] = ctr - 0.5f*w;
    sE[p] = ctr + 0.5f*w;
  }
  __syncthreads();

  const int lane = tid & 31;
  const int half = lane >> 4;   // 0: lanes 0-15, 1: lanes 16-31
  const int l16  = lane & 15;
  const int wv   = tid >> 5;    // wave id, 0..15

  // ---- blocked greedy NMS: 16 pivots per serial step ----
  for (int b0 = 0; b0 < M; b0 += 16) {
    for (int i = tid; i < 16*NWORDS; i += TPB) ((unsigned*)rowbits)[i] = 0u;
    __syncthreads();

    // per-lane constants for the 8 rows this lane owns in C/D layout
    float rS[8], rE[8];
    int rbase = b0 + 8*half;
    #pragma unroll
    for (int v = 0; v < 8; ++v) { rS[v] = sS[rbase+v]; rE[v] = sE[rbase+v]; }

    // A fragment (16x4): col0 = l_row, col1 = 1, cols 2-3 = 0
    v2f afrag;
    afrag.x = half ? 0.f : (sE[b0+l16] - sS[b0+l16]);
    afrag.y = half ? 0.f : 1.f;

    const int t0 = b0 >> 4;
    const int tLast = (M - 1) >> 4;
    for (int t = t0 + wv; t <= tLast; t += (TPB/32)) {
      const int c0 = t << 4;
      const int j  = c0 + l16;
      float cS = sS[j], cE = sE[j];
      // B fragment (4x16): row0 = 1, row1 = l_col, rows 2-3 = 0
      v2f bfrag;
      bfrag.x = half ? 0.f : 1.f;
      bfrag.y = half ? 0.f : (cE - cS);
      v8f cz = {};
      // P[r][j] = l_r + l_j via one matrix op per 16x16 tile
      v8f d = __builtin_amdgcn_wmma_f32_16x16x4_f32(
          false, afrag, false, bfrag, (short)0, cz, false, false);

      // 8 wave-uniform ballot masks, one per C/D VGPR row
      unsigned m[8];
      #pragma unroll
      for (int v = 0; v < 8; ++v) {
        int gr = b0 + v + 8*half;
        float q = fabsf(rE[v]-cE) + fabsf(rS[v]-cS);
        // IoU > 0.5  <=>  P > 3Q  (1-D boxes, positive lengths)
        bool cond = (j > gr) && (d[v] > 3.0f*q);
        m[v] = __builtin_amdgcn_ballot_w32(cond);
      }

      // distribute the 16 row-word ORs across 16 lanes:
      // lane k (k<8)   -> row k,    low 16 bits of m[k]
      // lane 16+k      -> row k+8, high 16 bits of m[k]
      unsigned sel = m[0];
      #pragma unroll
      for (int v = 1; v < 8; ++v) sel = (l16 == v) ? m[v] : sel;
      unsigned bits = half ? (sel >> 16) : (sel & 0xFFFFu);
      const unsigned w  = (unsigned)(c0 >> 5);
      const unsigned sh = (unsigned)(c0 & 31);   // 0 or 16
      if (l16 < 8) atomicOr(&rowbits[l16 + 8*half][w], bits << sh);
    }
    __syncthreads();

    // serial resolve of the 16 pivots (greedy order preserved)
    if (tid == 0) {
      unsigned act = 0, removed = 0;
      unsigned w0  = (unsigned)(b0 >> 5);
      unsigned sh0 = (unsigned)(b0 & 31);
      for (int r = 0; r < 16; ++r) {
        int gr = b0 + r;
        if (gr >= M) break;
        bool kb = (keepMask[gr >> 5] >> (gr & 31)) & 1u;
        if (kb && !((removed >> r) & 1u)) {
          act |= 1u << r;
          removed |= (rowbits[r][w0] >> sh0) & 0xFFFFu;
        }
      }
      sAct = act;
    }
    __syncthreads();

    // parallel apply of active pivots' suppression rows
    unsigned act = sAct;
    for (int w = tid; w < NWORDS; w += TPB) {
      unsigned acc = 0u, a2 = act;
      while (a2) {
        int r = __builtin_ctz(a2);
        a2 &= a2 - 1u;
        acc |= rowbits[r][w];
      }
      keepMask[w] &= ~acc;
    }
    __syncthreads();
  }

  // ---- scatter kept scores (recompute softmax score for kept entries) ----
  float* ob = out + (size_t)b*OUTSTRIDE + KEEPOFF + (size_t)c*NANCH;
  for (int p = tid; p < NANCH; p += TPB) {
    if (p < M && ((keepMask[p >> 5] >> (p & 31)) & 1u)) {
      int o = (int)sIdx[p];
      const float* z = cls + ((size_t)(b*NANCH + o)) * NCLS;
      ob[o] = fg_score(z, c);
    }
  }
}

extern "C" void kernel_launch(void* const* d_in, const int* in_sizes, int n_in,
                              void* d_out, int out_size, void* d_ws, size_t ws_size,
                              hipStream_t stream) {
  const float* loc  = (const float*)d_in[0];  // localizations (B,N,2)
  const float* cls  = (const float*)d_in[1];  // classifications (B,N,5)
  const float* defs = (const float*)d_in[2];  // localizations_default (N,2)
  float* out = (float*)d_out;                 // (B, 24576) f32

  hipLaunchKernelGGL(decode_zero_kernel, dim3((BATCH*NANCH + 255)/256), dim3(256),
                     0, stream, loc, defs, out);
  hipLaunchKernelGGL(nms_kernel, dim3(BATCH*4), dim3(TPB),
                     0, stream, loc, cls, defs, out);
}